// DifferentiableFSA_12532714570284
// MI455X (gfx1250) — compile-verified
//
#include <hip/hip_runtime.h>
#include <hip/hip_bf16.h>
#include <math.h>

typedef __attribute__((ext_vector_type(2))) float v2f;
typedef __attribute__((ext_vector_type(8))) float v8f;
typedef __attribute__((ext_vector_type(4))) unsigned int u32x4;
typedef __attribute__((ext_vector_type(8))) int i32x8;
typedef __attribute__((ext_vector_type(4))) int i32x4;

#define LSTR 72          // padded LDS row stride (floats); TDM pad cfg reproduces this
#define NEG_INF_F (-1e9f)

__global__ __launch_bounds__(128) void fsa_fused_kernel(
    const int*   __restrict__ tok,
    const float* __restrict__ emb,   // (21,64)
    const float* __restrict__ Wn,    // (4,64)
    const float* __restrict__ bn,    // (64)
    const float* __restrict__ Wm,    // (64,64)
    const float* __restrict__ bm,    // (64)
    const float* __restrict__ gamma, // (64)
    const float* __restrict__ beta,  // (64)
    const float* __restrict__ Ws,    // (64,1)
    const float* __restrict__ bs,    // (1)
    const float* __restrict__ Wo,    // (64,4)
    const float* __restrict__ bo,    // (4)
    const float* __restrict__ Wr,    // (2,64)
    const float* __restrict__ br,    // (64)
    float* __restrict__ out, int B)
{
  __shared__ float xb[64 * LSTR];   // x0 (pre-GEMM activations)
  __shared__ float yb[64 * LSTR];   // GEMM out -> gelu -> layernorm (in place)
  __shared__ float wmL[64 * LSTR];  // staged Wm (filled by TDM DMA)
  __shared__ float digv[64], isdig[64], mk[64];
  __shared__ float sc[64], wgt[64], lmA[64], rmA[64], pl[64];
  __shared__ float lrS[2];   // left_val, right_val
  __shared__ float lgS[4];   // op logits
  __shared__ float rvS[2];   // result_compressed, valid

  const int b = blockIdx.x;
  const int t = threadIdx.x;

  // ---- TDM: async DMA Wm (64x64 f32) global -> LDS with 8-DWORD row padding ----
  // D# per CDNA5 ISA ch.8. One wave issues (EXEC ignored by TENSOR ops).
  if (t < 32) {
    const unsigned long long ga = (unsigned long long)(uintptr_t)Wm; // 57-bit global addr
    const unsigned lds_off = (unsigned)(uintptr_t)(void*)wmL;        // low 32 bits = LDS byte addr

    u32x4 g0;
    g0[0] = 1u;                                  // count=1 valid descriptor
    g0[1] = lds_off;                             // lds_addr (bytes)
    g0[2] = (unsigned)(ga & 0xffffffffu);        // global_addr[31:0]
    g0[3] = (unsigned)((ga >> 32) & 0x1ffffffu)  // global_addr[56:32]
          | (2u << 30);                          // type=2 ("image")

    i32x8 g1;
    g1[0] = (int)((2u << 16)    // data_size: 2 -> 4 bytes
          | (1u << 20)          // pad_enable
          | (5u << 22)          // pad_interval: 2<<5 = 64 DWORDs (one row)
          | (7u << 25));        // pad_amount: 7+1 = 8 DWORDs  -> LDS stride 72
    g1[1] = (int)(64u << 16);   // tensor_dim0 = 64 (low 16 @ bits 63:48)
    g1[2] = (int)(64u << 16);   // tensor_dim0 hi=0 ; tensor_dim1 = 64 (low 16)
    g1[3] = (int)(64u << 16);   // tensor_dim1 hi=0 ; tile_dim0 = 64
    g1[4] = 64;                 // tile_dim1 = 64 ; tile_dim2 = 0
    g1[5] = 64;                 // tensor_dim0_stride = 64 (low 32)
    g1[6] = 0;                  // stride0 hi ; tensor_dim1_stride lo (unused, 2D)
    g1[7] = 0;

    i32x4 gz4 = {0, 0, 0, 0};           // D# groups 2/3: dims >= 3 unused
    i32x8 gz8 = {0, 0, 0, 0, 0, 0, 0, 0}; // extra group (6-arg form), zero-filled
    __builtin_amdgcn_tensor_load_to_lds(g0, g1, gz4, gz4, gz8, 0);
  }

  // ---- token features + x0 = emb[tok]*sqrt(H) + numeric@Wn + bn (overlaps DMA) ----
  {
    const int l  = t >> 1;
    const int hb = (t & 1) * 32;
    const int id = tok[b * 64 + l];
    const float fid  = (float)id;
    const float isd  = (id >= 4 && id <= 13) ? 1.f : 0.f;
    const float dv   = (fid - 4.f) * isd;
    const float opt  = (id == 14) ? 1.f : (id == 15) ? 2.f : (id == 16) ? 3.f : (id == 17) ? 4.f : 0.f;
    const float isop = (id >= 14 && id <= 17) ? 1.f : 0.f;
    if ((t & 1) == 0) { digv[l] = dv; isdig[l] = isd; mk[l] = (id != 0) ? 1.f : 0.f; }
    const float* e = emb + id * 64;
    for (int h = hb; h < hb + 32; ++h) {
      xb[l * LSTR + h] = e[h] * 8.f + dv * Wn[h] + isd * Wn[64 + h]
                       + opt * Wn[128 + h] + isop * Wn[192 + h] + bn[h];
    }
  }

  // TDM completion (wave 0's TENSORcnt), then publish LDS to all waves.
  if (t < 32) __builtin_amdgcn_s_wait_tensorcnt(0);
  __syncthreads();

  // ---- GEMM: y = x0 @ Wm + bm  via V_WMMA_F32_16X16X4_F32 ----
  // wave w owns M-strip [16w, 16w+16); loops 4 N-tiles; K=64 in steps of 4.
  {
    const int lane = t & 31;
    const int wv   = t >> 5;
    const int half = lane >> 4;   // which 16-lane half
    const int mr   = lane & 15;
    const int m0   = wv * 16;
    const float* arow = &xb[(m0 + mr) * LSTR];
    for (int nt = 0; nt < 4; ++nt) {
      const int n0 = nt * 16;
      v8f acc = {0.f, 0.f, 0.f, 0.f, 0.f, 0.f, 0.f, 0.f};
#pragma unroll
      for (int k0 = 0; k0 < 64; k0 += 4) {
        v2f a, bf;
        a[0]  = arow[k0 + 2 * half];
        a[1]  = arow[k0 + 2 * half + 1];
        bf[0] = wmL[(k0 + 2 * half) * LSTR + n0 + mr];
        bf[1] = wmL[(k0 + 2 * half + 1) * LSTR + n0 + mr];
        acc = __builtin_amdgcn_wmma_f32_16x16x4_f32(
            false, a, false, bf, (short)0, acc, false, false);
      }
#pragma unroll
      for (int r = 0; r < 8; ++r) {
        const int row = m0 + r + 8 * half;
        const int col = n0 + mr;
        yb[row * LSTR + col] = acc[r] + bm[col];
      }
    }
  }
  __syncthreads();

  // ---- per-row: exact gelu, layernorm, score ----
  if (t < 64) {
    float* r = &yb[t * LSTR];
    float s1 = 0.f, s2 = 0.f;
    for (int h = 0; h < 64; ++h) {
      float y = r[h];
      float g = 0.5f * y * (1.f + erff(y * 0.70710678118654752f));
      r[h] = g; s1 += g; s2 += g * g;
    }
    const float mu  = s1 * (1.f / 64.f);
    const float var = s2 * (1.f / 64.f) - mu * mu;
    const float rsd = rsqrtf(var + 1e-6f);
    float score = 0.f;
    for (int h = 0; h < 64; ++h) {
      float xn = (r[h] - mu) * rsd * gamma[h] + beta[h];
      r[h] = xn;
      score += xn * Ws[h];
    }
    sc[t] = score + bs[0] + (1.f - mk[t]) * NEG_INF_F;
  }
  __syncthreads();

  // ---- softmax over L ----
  if (t < 64) {
    float mx = sc[0];
    for (int j = 1; j < 64; ++j) mx = fmaxf(mx, sc[j]);
    float se = 0.f;
    for (int j = 0; j < 64; ++j) se += expf(sc[j] - mx);
    const float w = expf(sc[t] - mx) / se;
    wgt[t] = w;
    out[72 * B + b * 64 + t] = w;   // reduction_weights
  }
  __syncthreads();

  // ---- cumsum -> left/right masks ----
  if (t < 64) {
    float cum = 0.f;
    for (int j = 0; j <= t; ++j) cum += wgt[j];
    lmA[t] = (1.f - cum) * isdig[t];
    rmA[t] = (cum - wgt[t]) * isdig[t];
  }
  __syncthreads();

  // ---- soft digit assembly (thread 0: left, thread 1: right) ----
  if (t < 2) {
    const float* m = (t == 0) ? lmA : rmA;
    float total = 0.f;
    for (int j = 0; j < 64; ++j) total += m[j];
    float run = 0.f, val = 0.f;
    for (int j = 0; j < 64; ++j) {
      run += m[j];
      const float p = (total - run) * m[j];
      val += digv[j] * expf(p * 2.302585092994046f) * m[j];  // 10^p
    }
    lrS[t] = val;
    out[(66 + t) * B + b] = val;    // left_val, right_val
  }
  __syncthreads();

  // ---- weighted pooling over L ----
  if (t < 64) {
    float p = 0.f;
    for (int l = 0; l < 64; ++l) p += yb[l * LSTR + t] * wgt[l];
    pl[t] = p;
  }
  __syncthreads();

  // ---- op logits (H=64 -> 4) ----
  if (t < 4) {
    float lg = 0.f;
    for (int h = 0; h < 64; ++h) lg += pl[h] * Wo[h * 4 + t];
    lg += bo[t];
    lgS[t] = lg;
    out[68 * B + b * 4 + t] = lg;   // op_logits
  }
  __syncthreads();

  // ---- soft op-select + fixed ops ----
  if (t == 0) {
    float mx = fmaxf(fmaxf(lgS[0], lgS[1]), fmaxf(lgS[2], lgS[3]));
    float e0 = expf(lgS[0] - mx), e1 = expf(lgS[1] - mx);
    float e2 = expf(lgS[2] - mx), e3 = expf(lgS[3] - mx);
    const float inv = 1.f / (e0 + e1 + e2 + e3);
    e0 *= inv; e1 *= inv; e2 *= inv; e3 *= inv;
    const float lv = lrS[0], rv = lrS[1];
    const float dvf  = (fabsf(rv) > 1e-6f) ? 1.f : 0.f;
    const float safe = (dvf != 0.f) ? rv : 1.f;
    const float res = e0 * (lv + rv) + e1 * (lv - rv) + e2 * (lv * rv)
                    + e3 * ((lv / safe) * dvf);
    const float validv = 1.f - e3 * (1.f - dvf);
    out[b]     = res;       // result
    out[B + b] = validv;    // valid
    const float sgn = (res > 0.f) ? 1.f : ((res < 0.f) ? -1.f : 0.f);
    rvS[0] = sgn * log1pf(fabsf(res));
    rvS[1] = validv;
  }
  __syncthreads();

  // ---- result embedding: [compressed, valid] @ Wr + br ----
  if (t < 64) {
    out[2 * B + b * 64 + t] = rvS[0] * Wr[t] + rvS[1] * Wr[64 + t] + br[t];
  }
}

extern "C" void kernel_launch(void* const* d_in, const int* in_sizes, int n_in,
                              void* d_out, int out_size, void* d_ws, size_t ws_size,
                              hipStream_t stream) {
  const int*   tok   = (const int*)  d_in[0];
  const float* emb   = (const float*)d_in[1];
  const float* Wn    = (const float*)d_in[2];
  const float* bn    = (const float*)d_in[3];
  const float* Wm    = (const float*)d_in[4];
  const float* bm    = (const float*)d_in[5];
  const float* gamma = (const float*)d_in[6];
  const float* beta  = (const float*)d_in[7];
  const float* Ws    = (const float*)d_in[8];
  const float* bs    = (const float*)d_in[9];
  const float* Wo    = (const float*)d_in[10];
  const float* bo    = (const float*)d_in[11];
  const float* Wr    = (const float*)d_in[12];
  const float* br    = (const float*)d_in[13];
  const int B = in_sizes[0] / 64;

  fsa_fused_kernel<<<B, 128, 0, stream>>>(
      tok, emb, Wn, bn, Wm, bm, gamma, beta, Ws, bs, Wo, bo, Wr, br,
      (float*)d_out, B);
}